// DOC_Retriever_7516192768778
// MI455X (gfx1250) — compile-verified
//
#include <hip/hip_runtime.h>
#include <hip/hip_bf16.h>
#include <math.h>

typedef __attribute__((ext_vector_type(2))) float v2f;
typedef __attribute__((ext_vector_type(8))) float v8f;

#define S_CAP 16      // cap on num_search (actual value read from device scalar)
#define K_CAP 2048    // cap on #clusters held in LDS for top-S selection
#define M_CAP (S_CAP * 256)  // cap on S*NPC candidate scores in LDS
#define NTHREADS 256

// ---------------------------------------------------------------- helpers
__device__ __forceinline__ float block_reduce_sum(float v, float* sbuf) {
  int t = threadIdx.x;
  sbuf[t] = v;
  __syncthreads();
  for (int off = NTHREADS >> 1; off > 0; off >>= 1) {
    if (t < off) sbuf[t] += sbuf[t + off];
    __syncthreads();
  }
  float r = sbuf[0];
  __syncthreads();
  return r;
}

// ---------------------------------------------------------------- 1) normalize queries
__global__ void k_qnorm(const float* __restrict__ q, float* __restrict__ qn, int D) {
  __shared__ float sbuf[NTHREADS];
  int b = blockIdx.x, t = threadIdx.x;
  const float* row = q + (size_t)b * D;
  float ss = 0.f;
  for (int i = t; i < D; i += NTHREADS) { float v = row[i]; ss = fmaf(v, v, ss); }
  ss = block_reduce_sum(ss, sbuf);
  float sc = 1.0f / fmaxf(sqrtf(ss), 1e-12f);
  float* dst = qn + (size_t)b * D;
  for (int i = t; i < D; i += NTHREADS) dst[i] = row[i] * sc;
}

// ---------------------------------------------------------------- 2) reciprocal center norms
__global__ void k_cnorm(const float* __restrict__ centers, float* __restrict__ rcn, int D) {
  __shared__ float sbuf[NTHREADS];
  int c = blockIdx.x, t = threadIdx.x;
  const float* row = centers + (size_t)c * D;
  float ss = 0.f;
  for (int i = t; i < D; i += NTHREADS) { float v = row[i]; ss = fmaf(v, v, ss); }
  ss = block_reduce_sum(ss, sbuf);
  if (t == 0) rcn[c] = 1.0f / fmaxf(sqrtf(ss), 1e-12f);
}

// ---------------------------------------------------------------- 3) cos_c = qn @ centers^T * rcn   (WMMA f32 16x16x4)
// One wave (32 lanes) computes a 16(query) x 16(cluster) tile of the (B x K) score matrix.
__global__ void k_wmma_qc(const float* __restrict__ qn, const float* __restrict__ centers,
                          const float* __restrict__ rcn, float* __restrict__ cosc,
                          int K, int D) {
  int l  = threadIdx.x;          // 0..31, one wave32
  int tn = blockIdx.x;           // cluster tile
  int tm = blockIdx.y;           // query tile
  int m16   = l & 15;
  int khalf = (l >> 4) << 1;     // lanes 0-15 -> K{0,1}; lanes 16-31 -> K{2,3}

  int rowq = tm * 16 + m16;                          // A row (query)
  int coln = tn * 16 + m16;                          // B column (cluster)
  int colc = (coln < K) ? coln : (K - 1);            // clamp: avoids divergence; masked at store

  const float* arow = qn      + (size_t)rowq * D + khalf;
  const float* brow = centers + (size_t)colc * D + khalf;

  v8f acc = {};
#pragma unroll 4
  for (int kk = 0; kk < D; kk += 4) {
    v2f a = *(const v2f*)(arow + kk);   // A: 16x4 f32, 2 VGPRs/lane
    v2f b = *(const v2f*)(brow + kk);   // B: 4x16 f32, 2 VGPRs/lane
    acc = __builtin_amdgcn_wmma_f32_16x16x4_f32(
        /*neg_a=*/false, a, /*neg_b=*/false, b,
        /*c_mod=*/(short)0, acc, /*reuse_a=*/false, /*reuse_b=*/false);
  }

  if (coln < K) {
    float sc = rcn[coln];
    int rbase = tm * 16 + ((l >> 4) << 3);  // lanes 16-31 hold rows M+8
#pragma unroll
    for (int r = 0; r < 8; ++r)
      cosc[(size_t)(rbase + r) * K + coln] = acc[r] * sc;
  }
}

// ---------------------------------------------------------------- 4) top-S clusters per query (iterative argmax, first-index tie-break)
__global__ void k_topS(const float* __restrict__ cosc, const int* __restrict__ nsearch,
                       int* __restrict__ c_idx, int K) {
  __shared__ float sv[K_CAP];
  __shared__ float rv[NTHREADS];
  __shared__ int   ri[NTHREADS];
  int b = blockIdx.x, t = threadIdx.x;
  int S = *nsearch; if (S > S_CAP) S = S_CAP;
  int Kc = (K < K_CAP) ? K : K_CAP;
  for (int i = t; i < Kc; i += NTHREADS) sv[i] = cosc[(size_t)b * K + i];
  __syncthreads();
  for (int it = 0; it < S; ++it) {
    float best = -INFINITY; int bi = 0x7fffffff;
    for (int i = t; i < Kc; i += NTHREADS) {
      float v = sv[i];
      if (v > best) { best = v; bi = i; }
    }
    rv[t] = best; ri[t] = bi;
    __syncthreads();
    for (int off = NTHREADS >> 1; off > 0; off >>= 1) {
      if (t < off) {
        float v2 = rv[t + off]; int i2 = ri[t + off];
        if (v2 > rv[t] || (v2 == rv[t] && i2 < ri[t])) { rv[t] = v2; ri[t] = i2; }
      }
      __syncthreads();
    }
    if (t == 0) { c_idx[b * S_CAP + it] = ri[0]; sv[ri[0]] = -INFINITY; }
    __syncthreads();
  }
}

// ---------------------------------------------------------------- 5) exact cosine vs. candidates (HBM-bound; wave32 reductions)
__global__ void k_score(const float* __restrict__ qn, const float* __restrict__ data,
                        const int* __restrict__ c_idx, const int* __restrict__ nsearch,
                        float* __restrict__ cand_cos, int D, int NPC) {
  __shared__ float qs[1024];
  int b = blockIdx.x, s = blockIdx.y;
  int S = *nsearch; if (S > S_CAP) S = S_CAP;
  if (s >= S) return;
  int c = c_idx[b * S_CAP + s];
  const float* base = data + (size_t)c * NPC * D;
  int t = threadIdx.x;
  for (int i = t; i < D; i += NTHREADS) qs[i] = qn[(size_t)b * D + i];
  __syncthreads();

  int wave = t >> 5, lane = t & 31;          // 8 waves of 32 (wave32)
  for (int n = wave; n < NPC; n += 8) {
    const float* row = base + (size_t)n * D;
    __builtin_prefetch(row + 8 * (size_t)D, 0, 1);   // global_prefetch_b8 for next row of this wave
    float dot = 0.f, ss = 0.f;
    for (int j = lane; j < D; j += 32) {
      float v = row[j];
      dot = fmaf(v, qs[j], dot);
      ss  = fmaf(v, v, ss);
    }
#pragma unroll
    for (int off = 16; off > 0; off >>= 1) {
      dot += __shfl_xor(dot, off, 32);
      ss  += __shfl_xor(ss,  off, 32);
    }
    if (lane == 0) {
      float nr = fmaxf(sqrtf(ss), 1e-12f);
      cand_cos[((size_t)b * S_CAP + s) * NPC + n] = dot / nr;
    }
  }
}

// ---------------------------------------------------------------- 6) global top-k merge + gather embeddings
__global__ void k_merge(const float* __restrict__ cand_cos, const int* __restrict__ c_idx,
                        const int* __restrict__ clusted_idx, const float* __restrict__ data,
                        const int* __restrict__ nsearch,
                        float* __restrict__ out_idx, float* __restrict__ out_emb,
                        int NPC, int D, int TOPK) {
  __shared__ float sv[M_CAP];
  __shared__ float rv[NTHREADS];
  __shared__ int   ri[NTHREADS];
  __shared__ int   topi[64];
  int b = blockIdx.x, t = threadIdx.x;
  int S = *nsearch; if (S > S_CAP) S = S_CAP;
  int M = S * NPC; if (M > M_CAP) M = M_CAP;

  for (int i = t; i < M; i += NTHREADS) {
    int s = i / NPC, n = i - s * NPC;
    sv[i] = cand_cos[((size_t)b * S_CAP + s) * NPC + n];
  }
  __syncthreads();

  for (int it = 0; it < TOPK; ++it) {
    float best = -INFINITY; int bi = 0x7fffffff;
    for (int i = t; i < M; i += NTHREADS) {
      float v = sv[i];
      if (v > best) { best = v; bi = i; }
    }
    rv[t] = best; ri[t] = bi;
    __syncthreads();
    for (int off = NTHREADS >> 1; off > 0; off >>= 1) {
      if (t < off) {
        float v2 = rv[t + off]; int i2 = ri[t + off];
        if (v2 > rv[t] || (v2 == rv[t] && i2 < ri[t])) { rv[t] = v2; ri[t] = i2; }
      }
      __syncthreads();
    }
    if (t == 0) { topi[it] = ri[0]; sv[ri[0]] = -INFINITY; }
    __syncthreads();
  }

  for (int it = 0; it < TOPK; ++it) {
    int ti = topi[it];
    int s = ti / NPC, n = ti - s * NPC;
    int c = c_idx[b * S_CAP + s];
    if (t == 0) out_idx[(size_t)b * TOPK + it] = (float)clusted_idx[(size_t)c * NPC + n];
    const float* row = data + ((size_t)c * NPC + n) * D;
    float* dst = out_emb + ((size_t)b * TOPK + it) * D;
    for (int i = t; i < D; i += NTHREADS) dst[i] = row[i];
  }
}

// ---------------------------------------------------------------- host launcher
static inline size_t align256(size_t x) { return (x + 255) & ~(size_t)255; }

extern "C" void kernel_launch(void* const* d_in, const int* in_sizes, int n_in,
                              void* d_out, int out_size, void* d_ws, size_t ws_size,
                              hipStream_t stream) {
  const float* querys      = (const float*)d_in[0];
  const float* data        = (const float*)d_in[1];
  const float* centers     = (const float*)d_in[2];
  const int*   clusted_idx = (const int*)d_in[3];
  const int*   d_nsearch   = (const int*)d_in[5];   // device scalar num_search

  // Derive shapes from flat sizes (host arithmetic only — graph-capture safe):
  //   in_sizes: [B*D, K*NPC*D, K*D, K*NPC, 1, 1]
  long long sz_q  = in_sizes[0];
  long long sz_d  = in_sizes[1];
  long long sz_c  = in_sizes[2];
  long long sz_ci = in_sizes[3];
  int K    = (int)((sz_c * sz_ci) / sz_d);          // (K*D)(K*NPC)/(K*NPC*D)
  int D    = (int)(sz_c / K);
  int NPC  = (int)(sz_ci / K);
  int B    = (int)(sz_q / D);
  int TOPK = (int)((long long)out_size / ((long long)B * (D + 1)));  // out = B*k + B*k*D

  // Workspace carve-out
  char* ws = (char*)d_ws;
  float* qn   = (float*)ws;  ws += align256((size_t)B * D * sizeof(float));
  float* rcn  = (float*)ws;  ws += align256((size_t)K * sizeof(float));
  float* cosc = (float*)ws;  ws += align256((size_t)B * K * sizeof(float));
  int*   cidx = (int*)ws;    ws += align256((size_t)B * S_CAP * sizeof(int));
  float* cand = (float*)ws;  ws += align256((size_t)B * S_CAP * NPC * sizeof(float));
  (void)ws_size; (void)n_in;

  float* out     = (float*)d_out;
  float* out_idx = out;
  float* out_emb = out + (size_t)B * TOPK;

  k_qnorm<<<B, NTHREADS, 0, stream>>>(querys, qn, D);
  k_cnorm<<<K, NTHREADS, 0, stream>>>(centers, rcn, D);

  dim3 gw((K + 15) / 16, (B + 15) / 16);
  k_wmma_qc<<<gw, 32, 0, stream>>>(qn, centers, rcn, cosc, K, D);

  k_topS<<<B, NTHREADS, 0, stream>>>(cosc, d_nsearch, cidx, K);

  dim3 gs(B, S_CAP);
  k_score<<<gs, NTHREADS, 0, stream>>>(qn, data, cidx, d_nsearch, cand, D, NPC);

  k_merge<<<B, NTHREADS, 0, stream>>>(cand, cidx, clusted_idx, data, d_nsearch,
                                      out_idx, out_emb, NPC, D, TOPK);
}